// ModifiedALE_1176821039621
// MI455X (gfx1250) — compile-verified
//
#include <hip/hip_runtime.h>

// ---------------------------------------------------------------------------
// Graph message-passing (SIR-style) on MI455X / gfx1250.
//
// Bandwidth/atomic-bound scatter-gather. Edge stream (~77 MB) fits in the
// 192 MB L2 -> steps 1..9 re-hit L2. The edge stream is staged into LDS with
// gfx1250 global_load_async_to_lds_b128 (ASYNCcnt-tracked), double-buffered
// per wave so the async copy of tile t+1 overlaps gathers + f32 global
// atomics of tile t. Scatter uses the hardware no-return f32 atomic
// (global_atomic_add_f32, STOREcnt-tracked) emitted directly via inline asm.
// ---------------------------------------------------------------------------

constexpr int kBlock = 256;          // 8 wave32 waves per block
constexpr int kEdgeBlocks = 1024;    // 6250 tiles / 1024 blocks ~ 6 tiles each

// Hardware no-return f32 scatter-add (gfx1250 VGLOBAL atomic, TH=0/RT).
__device__ __forceinline__ void fatomic_add(float* addr, float val) {
  asm volatile("global_atomic_add_f32 %0, %1, off"
               :: "v"(addr), "v"(val) : "memory");
}

// ---- tail kernel (FIRST in file: small, so the disasm snippet shows the
//      full gather + global_atomic_add_f32 sequence). Launched only when
//      n_edges % 4 != 0 (never for the 6.4M-edge shape). --------------------
__global__ void sir_edges_tail(const int*   __restrict__ src,
                               const int*   __restrict__ dst,
                               const float* __restrict__ probs,
                               const float* __restrict__ cur,
                               float*       __restrict__ nxt,
                               const float* __restrict__ scales,
                               int step, int tail_start, int n_edges) {
  int e = tail_start + blockIdx.x * blockDim.x + threadIdx.x;
  if (e < n_edges) {
    fatomic_add(&nxt[dst[e]], cur[src[e]] * (probs[e] * scales[step]));
  }
}

// ---- edge kernel: nxt[dst[e]] += cur[src[e]] * probs[e] * scale -----------
__global__ void __launch_bounds__(kBlock)
sir_edges(const int*   __restrict__ src,
          const int*   __restrict__ dst,
          const float* __restrict__ probs,
          const float* __restrict__ cur,
          float*       __restrict__ nxt,
          const float* __restrict__ scales,
          int step, int n_edges) {
  __shared__ int4   lsrc[2][kBlock];
  __shared__ int4   ldst[2][kBlock];
  __shared__ float4 lprb[2][kBlock];

  const float scale  = scales[step];            // uniform -> scalar load
  const int n4       = n_edges >> 2;            // quad-edges (full quads only)
  const int ntiles   = (n4 + kBlock - 1) / kBlock;
  const int tid      = threadIdx.x;
  const int tstride  = gridDim.x;               // tiles advance by grid
  const int pfquads  = 2 * kEdgeBlocks * kBlock; // prefetch 2 strides ahead

  // Stage one tile (3 x b128 per lane) into LDS buffer b via the async path.
  auto issue = [&](int tile, int b) {
    const int q = tile * kBlock + tid;
    if (q < n4) {
      const int e = q << 2;
      // Warm L2 well ahead (gfx1250 global_prefetch_b8; speculative).
      if (q + pfquads < n4) {
        const int ep = (q + pfquads) << 2;
        __builtin_prefetch(src + ep, 0, 3);
        __builtin_prefetch(dst + ep, 0, 3);
        __builtin_prefetch(probs + ep, 0, 3);
      }
      const unsigned ls = (unsigned)(size_t)&lsrc[b][tid];
      const unsigned ld = (unsigned)(size_t)&ldst[b][tid];
      const unsigned lp = (unsigned)(size_t)&lprb[b][tid];
      asm volatile("global_load_async_to_lds_b128 %0, %1, off"
                   :: "v"(ls), "v"(src + e) : "memory");
      asm volatile("global_load_async_to_lds_b128 %0, %1, off"
                   :: "v"(ld), "v"(dst + e) : "memory");
      asm volatile("global_load_async_to_lds_b128 %0, %1, off"
                   :: "v"(lp), "v"(probs + e) : "memory");
    }
  };

  int tile = blockIdx.x;
  int buf  = 0;
  if (tile < ntiles) issue(tile, 0);

  while (tile < ntiles) {
    const int  next = tile + tstride;
    const bool hn   = next < ntiles;            // block-uniform
    if (hn) {
      issue(next, buf ^ 1);
      // 3 ops of tile(next) outstanding; async loads complete in order, so
      // waiting ASYNCcnt<=3 guarantees this tile's 3 staging copies landed.
      asm volatile("s_wait_asynccnt 0x3" ::: "memory");
    } else {
      asm volatile("s_wait_asynccnt 0x0" ::: "memory");
    }

    const int q = tile * kBlock + tid;
    if (q < n4) {
      const int4   s = lsrc[buf][tid];          // ds_load_b128
      const int4   d = ldst[buf][tid];
      const float4 p = lprb[buf][tid];
      fatomic_add(&nxt[d.x], cur[s.x] * (p.x * scale));
      fatomic_add(&nxt[d.y], cur[s.y] * (p.y * scale));
      fatomic_add(&nxt[d.z], cur[s.z] * (p.z * scale));
      fatomic_add(&nxt[d.w], cur[s.w] * (p.w * scale));
    }
    tile = next;
    buf ^= 1;
  }
}

// ---- init: cur = x, surv = 1 - x, nxt = 0 ---------------------------------
__global__ void sir_init(const float* __restrict__ x,
                         float* __restrict__ cur,
                         float* __restrict__ nxt,
                         float* __restrict__ surv,
                         int n) {
  int i = blockIdx.x * blockDim.x + threadIdx.x;
  if (i < n) {
    float v = x[i];
    cur[i]  = v;
    surv[i] = 1.0f - v;
    nxt[i]  = 0.0f;
  }
}

// ---- per-step scalar scales: edge_weight[i] * exp(-time_decay[i]^2) -------
__global__ void sir_scales(const float* __restrict__ time_decay,
                           const float* __restrict__ edge_weight,
                           float* __restrict__ scales,
                           int steps) {
  int i = threadIdx.x;
  if (i < steps) {
    float t = time_decay[i];
    scales[i] = edge_weight[i] * expf(-t * t);
  }
}

// ---- node update: cur = nxt + bias; surv *= (1 - cur); nxt = 0 ------------
__global__ void sir_node(float* __restrict__ cur,
                         float* __restrict__ nxt,
                         float* __restrict__ surv,
                         const float* __restrict__ bias,
                         int n) {
  int i = blockIdx.x * blockDim.x + threadIdx.x;
  if (i < n) {
    float v = nxt[i] + bias[0];
    surv[i] *= (1.0f - v);
    cur[i]  = v;
    nxt[i]  = 0.0f;    // self-resetting accumulator: graph replay is clean
  }
}

// ---- finalize: out = clip(1 - surv, 0, 1) ---------------------------------
__global__ void sir_final(const float* __restrict__ surv,
                          float* __restrict__ out,
                          int n) {
  int i = blockIdx.x * blockDim.x + threadIdx.x;
  if (i < n) {
    float f = 1.0f - surv[i];
    out[i] = fminf(fmaxf(f, 0.0f), 1.0f);
  }
}

// ---------------------------------------------------------------------------
extern "C" void kernel_launch(void* const* d_in, const int* in_sizes, int n_in,
                              void* d_out, int out_size, void* d_ws, size_t ws_size,
                              hipStream_t stream) {
  // setup_inputs() order:
  //   0: x           (N,1)  f32
  //   1: edge_index  (2,E)  int (row 0 = src, row 1 = dst)
  //   2: edge_probs  (E,)   f32
  //   3: time_decay  (S,)   f32
  //   4: node_bias   scalar f32
  //   5: edge_weight (S,)   f32
  const float* x           = (const float*)d_in[0];
  const int*   edge_index  = (const int*)  d_in[1];
  const float* edge_probs  = (const float*)d_in[2];
  const float* time_decay  = (const float*)d_in[3];
  const float* node_bias   = (const float*)d_in[4];
  const float* edge_weight = (const float*)d_in[5];

  const int n_nodes = in_sizes[0];  // 100000
  const int n_edges = in_sizes[2];  // 6400000
  const int n_steps = in_sizes[3];  // 10

  const int* src = edge_index;
  const int* dst = edge_index + n_edges;

  // Workspace: cur | nxt | surv | scales   (~1.2 MB of f32)
  float* ws     = (float*)d_ws;
  float* cur    = ws;
  float* nxt    = ws + (size_t)n_nodes;
  float* surv   = ws + (size_t)2 * n_nodes;
  float* scales = ws + (size_t)3 * n_nodes;

  const int nodeBlocks = (n_nodes + kBlock - 1) / kBlock;
  const int tail_start = n_edges & ~3;

  sir_init  <<<nodeBlocks, kBlock, 0, stream>>>(x, cur, nxt, surv, n_nodes);
  sir_scales<<<1, 32, 0, stream>>>(time_decay, edge_weight, scales, n_steps);

  for (int s = 0; s < n_steps; ++s) {
    sir_edges<<<kEdgeBlocks, kBlock, 0, stream>>>(src, dst, edge_probs,
                                                  cur, nxt, scales, s, n_edges);
    if (tail_start < n_edges) {
      sir_edges_tail<<<1, 128, 0, stream>>>(src, dst, edge_probs,
                                            cur, nxt, scales, s,
                                            tail_start, n_edges);
    }
    sir_node <<<nodeBlocks, kBlock, 0, stream>>>(cur, nxt, surv, node_bias, n_nodes);
  }

  sir_final<<<nodeBlocks, kBlock, 0, stream>>>(surv, (float*)d_out, n_nodes);
}